// CoordinateDescent_88871463289191
// MI455X (gfx1250) — compile-verified
//
#include <hip/hip_runtime.h>

typedef __attribute__((ext_vector_type(2))) float v2f;
typedef __attribute__((ext_vector_type(4))) float f4;
typedef __attribute__((ext_vector_type(8))) float v8f;

#define BB 4
#define MM 4096
#define NN 4096
#define RR 16
#define EPSF 1e-8f

// ---------------------------------------------------------------------------
// Gram: G[b] = W[b]^T @ W[b]   (W: (B, N, R) row-major, G: (B, R, R))
// One block per batch, 256 threads = one thread per (r1, r2) entry.
// W streamed through LDS in 128-row chunks (8 KB).
// ---------------------------------------------------------------------------
__global__ __launch_bounds__(256) void gram_kernel(const float* __restrict__ W,
                                                   float* __restrict__ G) {
    __shared__ float lds[128 * RR];
    const int b   = blockIdx.x;
    const int tid = threadIdx.x;
    const int r1  = tid >> 4;
    const int r2  = tid & 15;
    const float* Wb = W + (size_t)b * NN * RR;
    float acc = 0.f;
    for (int nc = 0; nc < NN; nc += 128) {
        const f4* src = (const f4*)(Wb + (size_t)nc * RR);   // 2048 floats = 512 f4
        f4* dst = (f4*)lds;
        dst[tid]       = src[tid];
        dst[tid + 256] = src[tid + 256];
        __syncthreads();
#pragma unroll 8
        for (int n = 0; n < 128; ++n)
            acc = fmaf(lds[n * RR + r1], lds[n * RR + r2], acc);
        __syncthreads();
    }
    G[(size_t)b * RR * RR + tid] = acc;
}

// ---------------------------------------------------------------------------
// A[b,i,r] = sum_k Xe[i,k] * W[b,k,r]
//   TRANS=false: Xe[i,k] = X[b,i,k]   (a = x @ v)
//   TRANS=true : Xe[i,k] = X[b,k,i]   (a = x^T @ u)
// fp32 WMMA 16x16x4. 128 threads = 4 waves; each wave owns one 16-row tile
// (64 rows/block). W chunk (256 x 16 fp32 = 16 KB) staged in LDS, shared by
// all 4 waves. x is pure streaming (each element consumed exactly once) ->
// nontemporal loads so it never displaces reused data in WGP$/L2.
// EXEC all-ones: no divergence, exact grid.
// ---------------------------------------------------------------------------
template <bool TRANS>
__global__ __launch_bounds__(128) void xw_gemm_kernel(const float* __restrict__ X,
                                                      const float* __restrict__ W,
                                                      float* __restrict__ A) {
    constexpr int KC = 256;
    __shared__ float ldsW[KC * RR];           // 16 KB
    const int tid   = threadIdx.x;
    const int wave  = tid >> 5;
    const int lane  = tid & 31;
    const int tiles = MM / 64;                // blocks per batch
    const int b     = blockIdx.x / tiles;
    const int m0    = (blockIdx.x % tiles) * 64 + wave * 16;

    const float* Xb = X + (size_t)b * MM * NN;
    const float* Wb = W + (size_t)b * NN * RR;

    const int row   = m0 + (lane & 15);       // A-frag: M = lane%16
    const int khalf = lane >> 4;              // A/B-frag: K-pair = (lane/16)*2
    const int rcol  = lane & 15;              // B-frag: N(=R) = lane%16

    v8f c = {0.f, 0.f, 0.f, 0.f, 0.f, 0.f, 0.f, 0.f};

    for (int kc = 0; kc < NN; kc += KC) {
        // cooperative stage of W chunk: 4096 floats = 1024 f4 / 128 threads
        {
            const f4* src = (const f4*)(Wb + (size_t)kc * RR);
            f4* dst = (f4*)ldsW;
#pragma unroll
            for (int i = 0; i < 8; ++i)
                dst[tid + i * 128] = src[tid + i * 128];
        }
        __syncthreads();

#pragma unroll 8
        for (int k = 0; k < KC; k += 4) {
            const int kk = k + khalf * 2;     // local K of this lane's pair
            const int kg = kc + kk;           // global K
            v2f a;
            if (TRANS) {
                // Xe[row, kg] = X[kg, row]; lanes 0-15 are consecutive rows
                // -> coalesced 64B per K-row. Streaming: NT hint.
                a.x = __builtin_nontemporal_load(Xb + (size_t)(kg + 0) * MM + row);
                a.y = __builtin_nontemporal_load(Xb + (size_t)(kg + 1) * MM + row);
            } else {
                // contiguous K-pair within the lane's row. Streaming: NT hint.
                a = __builtin_nontemporal_load(
                        (const v2f*)(Xb + (size_t)row * NN + kg));
            }
            v2f bf;
            bf.x = ldsW[(kk + 0) * RR + rcol];
            bf.y = ldsW[(kk + 1) * RR + rcol];
            // D = A(16x4 f32) * B(4x16 f32) + C(16x16 f32)
            c = __builtin_amdgcn_wmma_f32_16x16x4_f32(
                    /*neg_a=*/false, a, /*neg_b=*/false, bf,
                    /*c_mod=*/(short)0, c, /*reuse_a=*/false, /*reuse_b=*/false);
        }
        __syncthreads();
    }

    // D layout: VGPR j -> M = m0 + j + 8*(lane/16), N = lane%16
    float* Ab = A + (size_t)b * MM * RR;
#pragma unroll
    for (int j = 0; j < 8; ++j)
        Ab[(size_t)(m0 + j + 8 * khalf) * RR + rcol] = c[j];
}

// ---------------------------------------------------------------------------
// Sequential coordinate-descent sweep, one thread per row:
//   for r: u[r] = (a[r] - (u.b[:,r] - u[r]*b[r,r]) + eps) / (b[r,r] + eps)
// Gram matrix in LDS (all threads of a block share one batch: MM % 256 == 0).
// ---------------------------------------------------------------------------
__global__ __launch_bounds__(256) void cd_kernel(const float* __restrict__ A,
                                                 const float* __restrict__ G,
                                                 const float* __restrict__ Uin,
                                                 float* __restrict__ Uout) {
    __shared__ float g[RR * RR];
    const int gid = blockIdx.x * 256 + threadIdx.x;   // global row index
    const int b   = gid / MM;
    g[threadIdx.x] = G[(size_t)b * RR * RR + threadIdx.x];
    __syncthreads();

    float ar[RR], ur[RR];
    const float* Ar = A + (size_t)gid * RR;
    const float* Ur = Uin + (size_t)gid * RR;
#pragma unroll
    for (int r = 0; r < RR; ++r) { ar[r] = Ar[r]; ur[r] = Ur[r]; }

#pragma unroll
    for (int r = 0; r < RR; ++r) {
        const float brr = g[r * RR + r];
        float cross = 0.f;
#pragma unroll
        for (int j = 0; j < RR; ++j)
            cross = fmaf(ur[j], g[j * RR + r], cross);
        cross -= ur[r] * brr;                 // exclude diagonal term
        ur[r] = (ar[r] - cross + EPSF) / (brr + EPSF);
    }

    float* Or = Uout + (size_t)gid * RR;
#pragma unroll
    for (int r = 0; r < RR; ++r) Or[r] = ur[r];
}

// ---------------------------------------------------------------------------
extern "C" void kernel_launch(void* const* d_in, const int* in_sizes, int n_in,
                              void* d_out, int out_size, void* d_ws, size_t ws_size,
                              hipStream_t stream) {
    const float* x = (const float*)d_in[0];   // (B, M, N)
    const float* u = (const float*)d_in[1];   // (B, M, R)
    const float* v = (const float*)d_in[2];   // (B, N, R)

    float* u_new = (float*)d_out;                       // (B, M, R)
    float* v_new = u_new + (size_t)BB * MM * RR;        // (B, N, R)

    float* ws_a = (float*)d_ws;                         // (B, M, R) scratch
    float* ws_g = ws_a + (size_t)BB * MM * RR;          // (B, R, R) scratch

    const int gemm_blocks = BB * (MM / 64);
    const int cd_blocks   = BB * MM / 256;

    // ---- u update ----
    gram_kernel<<<BB, 256, 0, stream>>>(v, ws_g);                       // G = v^T v
    xw_gemm_kernel<false><<<gemm_blocks, 128, 0, stream>>>(x, v, ws_a); // a = x @ v
    cd_kernel<<<cd_blocks, 256, 0, stream>>>(ws_a, ws_g, u, u_new);     // u_new

    // ---- v update (uses u_new) ----
    gram_kernel<<<BB, 256, 0, stream>>>(u_new, ws_g);                        // G = u_new^T u_new
    xw_gemm_kernel<true><<<gemm_blocks, 128, 0, stream>>>(x, u_new, ws_a);   // a = x^T @ u_new
    cd_kernel<<<cd_blocks, 256, 0, stream>>>(ws_a, ws_g, v, v_new);          // v_new
}